// Denoise_17566416241425
// MI455X (gfx1250) — compile-verified
//
#include <hip/hip_runtime.h>
#include <math.h>

#define U_N 100000
#define I_N 50000
#define D_N 64
#define K3  192

typedef float v2f __attribute__((ext_vector_type(2)));
typedef float v8f __attribute__((ext_vector_type(8)));
typedef unsigned int v4u __attribute__((ext_vector_type(4)));
typedef int v4i __attribute__((ext_vector_type(4)));
typedef int v8i __attribute__((ext_vector_type(8)));

// ---------------------------------------------------------------- TDM helper
// 2D tensor load Global -> LDS via the Tensor Data Mover.
// tile_w dwords per row, tile_h rows, row stride `stride` dwords in memory.
// Pad 1 DWORD after every 64 DWORDs stored -> LDS row stride = tile_w + tile_w/64.
__device__ __forceinline__ void tdm_load_2d(unsigned lds_byte_addr, const void* gptr,
                                            unsigned tile_w, unsigned tile_h,
                                            unsigned stride) {
    unsigned long long ga = (unsigned long long)gptr;
    v4u g0;
    g0.x = 1u;                                            // count=1 (valid user D#)
    g0.y = lds_byte_addr;                                 // [63:32] lds_addr
    g0.z = (unsigned)(ga & 0xffffffffu);                  // global_addr[31:0]
    g0.w = (unsigned)((ga >> 32) & 0x1ffffffu) | (2u << 30); // addr[56:32] | type=2
    v8i g1;
    g1[0] = (int)((2u << 16)      // data_size = 4 bytes
                | (1u << 20)      // pad_enable
                | (5u << 22));    // pad_interval code 5 = every 64 DWORDs; pad_amount=0 -> 1 DWORD
    g1[1] = (int)((tile_w & 0xffffu) << 16);              // tensor_dim0[15:0] (barrier addr = 0)
    g1[2] = (int)(((tile_w >> 16) & 0xffffu) | ((tile_h & 0xffffu) << 16)); // dim0 hi | dim1 lo
    g1[3] = (int)(((tile_h >> 16) & 0xffffu) | ((tile_w & 0xffffu) << 16)); // dim1 hi | tile_dim0
    g1[4] = (int)(tile_h & 0xffffu);                      // tile_dim1 (tile_dim2 = 0)
    g1[5] = (int)stride;                                  // tensor_dim0_stride[31:0]
    g1[6] = 0;                                            // stride hi | dim1_stride lo
    g1[7] = 0;
    v4i gz4 = {0, 0, 0, 0};
    v8i gz8 = {0, 0, 0, 0, 0, 0, 0, 0};
    __builtin_amdgcn_tensor_load_to_lds(g0, g1, gz4, gz4, gz8, 0);
}

// ---------------------------------------------------------------- init
__global__ void init_kernel(const float* __restrict__ ue, const float* __restrict__ ie,
                            float* __restrict__ ego, float* __restrict__ soc,
                            float* __restrict__ acc) {
    size_t idx = (size_t)blockIdx.x * blockDim.x + threadIdx.x;
    const size_t nu = (size_t)U_N * D_N;
    const size_t nt = (size_t)(U_N + I_N) * D_N;
    if (idx >= nt) return;
    float v = (idx < nu) ? ue[idx] : ie[idx - nu];
    ego[idx] = v;
    acc[idx] = v;
    if (idx < nu) soc[idx] = v;
}

// ---------------------------------------------------------------- SpMM (COO, atomic scatter; gathers are L2-resident)
__global__ void spmm_kernel(const int* __restrict__ rows, const int* __restrict__ cols,
                            const float* __restrict__ vals, int nnz,
                            const float* __restrict__ x, float* __restrict__ y) {
    size_t t = (size_t)blockIdx.x * blockDim.x + threadIdx.x;
    int e = (int)(t >> 4);
    if (e >= nnz) return;
    int c4 = ((int)t & 15) << 2;
    int r = rows[e];
    int c = cols[e];
    float v = vals[e];
    const float4 xv = *(const float4*)(x + (size_t)c * D_N + c4);
    float* dst = y + (size_t)r * D_N + c4;
    unsafeAtomicAdd(dst + 0, v * xv.x);
    unsafeAtomicAdd(dst + 1, v * xv.y);
    unsafeAtomicAdd(dst + 2, v * xv.z);
    unsafeAtomicAdd(dst + 3, v * xv.w);
}

// ---------------------------------------------------------------- fused GEMM (+bias, +mish), WMMA f32 16x16x4
// out[M x N] = act( A[M x 192] @ W[N x 192]^T + bias )
// A staged into LDS by the Tensor Data Mover:
//   cat_mode=0: one TDM, 192x32 tile, padded -> row_stride 195, seg_stride 65
//   cat_mode=1: two TDMs (ue, soc planes of 32x65), product plane computed in LDS;
//               row_stride 65, seg_stride 2080
__global__ __launch_bounds__(256) void fusion_gemm_kernel(
        const float* __restrict__ Au, const float* __restrict__ As,
        const float* __restrict__ W, const float* __restrict__ bias,
        float* __restrict__ out, int M, int N, int apply_mish, int cat_mode) {
    __shared__ float lds[6240];            // 3 * 32 * 65 = 32 * 195 dwords = 24,960 B
    const int tid = threadIdx.x;
    const int wave = tid >> 5;
    const int row_base = blockIdx.x * 32;
    const unsigned lbase = (unsigned)(size_t)&lds[0];
    __builtin_prefetch(W, 0, 3);

    int seg_stride, row_stride;
    if (cat_mode) {
        seg_stride = 2080;                 // 32 * 65
        row_stride = 65;
        if (wave == 0) {
            tdm_load_2d(lbase,            Au + (size_t)row_base * D_N, 64, 32, 64);
            tdm_load_2d(lbase + 2080u*4u, As + (size_t)row_base * D_N, 64, 32, 64);
            __builtin_amdgcn_s_wait_tensorcnt(0);
        }
        __syncthreads();
        // product plane: cat = [ue | soc | ue*soc]
        for (int i = tid; i < 32 * 64; i += 256) {
            int r = i >> 6;
            int k = i & 63;
            lds[2 * 2080 + r * 65 + k] = lds[r * 65 + k] * lds[2080 + r * 65 + k];
        }
        __syncthreads();
    } else {
        seg_stride = 65;                   // pad inserts 1 dword per 64 -> (r,k) at r*195 + (k/64)*65 + k%64
        row_stride = 195;
        if (wave == 0) {
            tdm_load_2d(lbase, Au + (size_t)row_base * K3, 192, 32, 192);
            __builtin_amdgcn_s_wait_tensorcnt(0);
        }
        __syncthreads();
    }

    const int lane = tid & 31;
    const int half = lane >> 4;            // lanes 16-31 carry K+2/K+3 of the A fragment
    const int l16  = lane & 15;
    const int mt   = wave & 1;             // 16-row half of the 32-row tile
    const int ng   = wave >> 1;            // 0..3 strided ownership of N tiles
    const int ntiles = N >> 4;

    const float* arow = &lds[(mt * 16 + l16) * row_stride + 2 * half];

    for (int t = ng; t < ntiles; t += 4) {
        const int n = t * 16 + l16;
        const float* wrow = W + (size_t)n * K3 + 2 * half;
        v8f c = {};

        // 48-step K chain, software-pipelined with a 4-deep rolling prefetch
        v2f abuf[4], bbuf[4];
#pragma unroll
        for (int i = 0; i < 4; ++i) {
            const int k = i * 4;
            const float* ap = arow + (k >> 6) * seg_stride + (k & 63);
            abuf[i].x = ap[0];     abuf[i].y = ap[1];
            bbuf[i].x = wrow[k];   bbuf[i].y = wrow[k + 1];
        }
#pragma unroll
        for (int kk = 0; kk < 48; ++kk) {
            v2f a = abuf[kk & 3];
            v2f b = bbuf[kk & 3];
            if (kk + 4 < 48) {
                const int k = (kk + 4) * 4;
                const float* ap = arow + (k >> 6) * seg_stride + (k & 63);
                abuf[kk & 3].x = ap[0];     abuf[kk & 3].y = ap[1];
                bbuf[kk & 3].x = wrow[k];   bbuf[kk & 3].y = wrow[k + 1];
            }
            c = __builtin_amdgcn_wmma_f32_16x16x4_f32(
                    false, a, false, b, (short)0, c, false, false);
        }

        float bn = bias[n];
#pragma unroll
        for (int j = 0; j < 8; ++j) {
            int row = row_base + mt * 16 + half * 8 + j;  // C VGPR j: lanes<16 -> M=j, lanes>=16 -> M=j+8
            float x = c[j] + bn;
            if (apply_mish) {
                float sp = log1pf(expf(x));   // softplus
                x = x * tanhf(sp);            // mish
            }
            if (row < M) out[(size_t)row * N + t * 16 + l16] = x;
        }
    }
}

// ---------------------------------------------------------------- sum of squares (global Frobenius norm)
__global__ void sumsq_kernel(const float* __restrict__ x, int n, float* __restrict__ outsum) {
    __shared__ float sm[256];
    float v = 0.f;
    size_t stride = (size_t)gridDim.x * blockDim.x;
    for (size_t i = (size_t)blockIdx.x * blockDim.x + threadIdx.x; i < (size_t)n; i += stride) {
        float a = x[i];
        v += a * a;
    }
    sm[threadIdx.x] = v;
    __syncthreads();
    for (int s = 128; s > 0; s >>= 1) {
        if (threadIdx.x < s) sm[threadIdx.x] += sm[threadIdx.x + s];
        __syncthreads();
    }
    if (threadIdx.x == 0) unsafeAtomicAdd(outsum, sm[0]);
}

// ---------------------------------------------------------------- normalize t3, splice into ego, accumulate
__global__ void finalize_kernel(const float* __restrict__ t3, const float* __restrict__ sumsq,
                                float* __restrict__ ego, float* __restrict__ soc,
                                float* __restrict__ acc) {
    size_t idx = (size_t)blockIdx.x * blockDim.x + threadIdx.x;
    const size_t nu = (size_t)U_N * D_N;
    const size_t nt = (size_t)(U_N + I_N) * D_N;
    if (idx >= nt) return;
    float v;
    if (idx < nu) {
        float inv = rsqrtf(*sumsq);
        v = t3[idx] * inv;
        ego[idx] = v;
        soc[idx] = v;
    } else {
        v = ego[idx];
    }
    acc[idx] += v;
}

// ---------------------------------------------------------------- mean over the 4 collected embeddings
__global__ void mean_kernel(const float* __restrict__ acc, float* __restrict__ out) {
    size_t idx = (size_t)blockIdx.x * blockDim.x + threadIdx.x;
    const size_t nt = (size_t)(U_N + I_N) * D_N;
    if (idx >= nt) return;
    out[idx] = acc[idx] * 0.25f;
}

// ================================================================ host
extern "C" void kernel_launch(void* const* d_in, const int* in_sizes, int n_in,
                              void* d_out, int out_size, void* d_ws, size_t ws_size,
                              hipStream_t stream) {
    const float* user_emb = (const float*)d_in[0];
    const float* item_emb = (const float*)d_in[1];
    const int*   A_rows   = (const int*)d_in[2];
    const int*   A_cols   = (const int*)d_in[3];
    const float* A_vals   = (const float*)d_in[4];
    const int*   S_rows   = (const int*)d_in[5];
    const int*   S_cols   = (const int*)d_in[6];
    const float* S_vals   = (const float*)d_in[7];
    const float* W1 = (const float*)d_in[8];
    const float* b1 = (const float*)d_in[9];
    const float* W2 = (const float*)d_in[10];
    const float* b2 = (const float*)d_in[11];
    const float* W3 = (const float*)d_in[12];
    const float* b3 = (const float*)d_in[13];

    const int annz = in_sizes[2];
    const int snnz = in_sizes[5];

    const size_t NE  = (size_t)(U_N + I_N) * D_N;   // 9.6M
    const size_t NU  = (size_t)U_N * D_N;           // 6.4M
    const size_t NU3 = (size_t)U_N * K3;            // 19.2M

    float* ws    = (float*)d_ws;
    float* ego_a = ws;
    float* ego_b = ego_a + NE;
    float* accb  = ego_b + NE;
    float* soc_a = accb + NE;
    float* soc_b = soc_a + NU;
    float* t1    = soc_b + NU;
    float* t2    = t1 + NU3;
    float* t3    = t2 + NU3;
    float* scal  = t3 + NU;

    const int blocksNE = (int)((NE + 255) / 256);
    init_kernel<<<blocksNE, 256, 0, stream>>>(user_emb, item_emb, ego_a, soc_a, accb);

    float* ego_cur = ego_a; float* ego_nxt = ego_b;
    float* soc_cur = soc_a; float* soc_nxt = soc_b;

    const int fblocks = (U_N + 31) / 32;   // 3125

    for (int layer = 0; layer < 3; ++layer) {
        (void)hipMemsetAsync(ego_nxt, 0, NE * sizeof(float), stream);
        (void)hipMemsetAsync(soc_nxt, 0, NU * sizeof(float), stream);
        (void)hipMemsetAsync(scal, 0, sizeof(float), stream);

        {
            size_t tA = (size_t)annz * 16;
            spmm_kernel<<<(int)((tA + 255) / 256), 256, 0, stream>>>(
                A_rows, A_cols, A_vals, annz, ego_cur, ego_nxt);
            size_t tS = (size_t)snnz * 16;
            spmm_kernel<<<(int)((tS + 255) / 256), 256, 0, stream>>>(
                S_rows, S_cols, S_vals, snnz, soc_cur, soc_nxt);
        }

        // fusion MLP: t1 = mish(cat @ W1^T + b1) ; t2 = mish(t1 @ W2^T + b2) ; t3 = t2 @ W3^T + b3
        fusion_gemm_kernel<<<fblocks, 256, 0, stream>>>(ego_nxt, soc_nxt, W1, b1, t1, U_N, 192, 1, 1);
        fusion_gemm_kernel<<<fblocks, 256, 0, stream>>>(t1, nullptr,      W2, b2, t2, U_N, 192, 1, 0);
        fusion_gemm_kernel<<<fblocks, 256, 0, stream>>>(t2, nullptr,      W3, b3, t3, U_N, 64, 0, 0);

        sumsq_kernel<<<1024, 256, 0, stream>>>(t3, (int)NU, scal);
        finalize_kernel<<<blocksNE, 256, 0, stream>>>(t3, scal, ego_nxt, soc_nxt, accb);

        float* tmp = ego_cur; ego_cur = ego_nxt; ego_nxt = tmp;
        tmp = soc_cur; soc_cur = soc_nxt; soc_nxt = tmp;
    }

    mean_kernel<<<blocksNE, 256, 0, stream>>>(accb, (float*)d_out);
}